// Hyperplane_36945308680256
// MI455X (gfx1250) — compile-verified
//
#include <hip/hip_runtime.h>
#include <math.h>

// Problem constants (match reference).
constexpr int Bdim = 512;
constexpr int Kdim = 1024;
constexpr int Ddim = 256;
constexpr float EPS = 1e-6f;

// Tiling for the fused dual-GEMM kernel.
constexpr int TM = 32;        // batch rows per workgroup tile
constexpr int TN = 64;        // k columns per workgroup tile
constexpr int DC = 64;        // D-chunk staged in LDS per iteration
constexpr int LSTR = DC + 4;  // LDS row stride (floats); +4 pad -> conflict-free ds_load_b64

typedef float v2f __attribute__((ext_vector_type(2)));
typedef float v8f __attribute__((ext_vector_type(8)));

// ---------------------------------------------------------------------------
// Kernel 1: per-row stats.
//   blocks [0, K)      : row k of p,a  -> p2, pa, na = (1-p2)*||a||, lam*na
//   blocks [K, K+B)    : row b of x    -> x2
// ws layout (floats): p2[K] | pa[K] | na[K] | lamna[K] | x2[B]
// ---------------------------------------------------------------------------
__global__ __launch_bounds__(Ddim) void hyp_stats_kernel(
    const float* __restrict__ x, const float* __restrict__ p,
    const float* __restrict__ a, float* __restrict__ ws) {
  __shared__ float s0[Ddim];
  __shared__ float s1[Ddim];
  __shared__ float s2[Ddim];
  const int t = threadIdx.x;
  const int blk = blockIdx.x;

  if (blk < Kdim) {
    const float pv = p[(size_t)blk * Ddim + t];
    const float av = a[(size_t)blk * Ddim + t];
    s0[t] = pv * pv;
    s1[t] = pv * av;
    s2[t] = av * av;
    __syncthreads();
#pragma unroll
    for (int off = Ddim / 2; off > 0; off >>= 1) {
      if (t < off) {
        s0[t] += s0[t + off];
        s1[t] += s1[t + off];
        s2[t] += s2[t + off];
      }
      __syncthreads();
    }
    if (t == 0) {
      const float p2 = s0[0];
      const float pa = s1[0];
      const float a2 = s2[0];
      const float tt = 1.0f - p2;             // parallel-transport scale
      const float na = tt * sqrtf(a2);        // ||a_pt||
      ws[blk] = p2;
      ws[Kdim + blk] = pa;
      ws[2 * Kdim + blk] = na;
      ws[3 * Kdim + blk] = (2.0f / (tt + EPS)) * na;  // lam_p * ||a_pt||
    }
  } else {
    const int b = blk - Kdim;
    const float xv = x[(size_t)b * Ddim + t];
    s0[t] = xv * xv;
    __syncthreads();
#pragma unroll
    for (int off = Ddim / 2; off > 0; off >>= 1) {
      if (t < off) s0[t] += s0[t + off];
      __syncthreads();
    }
    if (t == 0) ws[4 * Kdim + b] = s0[0];
  }
}

// ---------------------------------------------------------------------------
// Kernel 2: fused dual GEMM (PX = X P^T, XA = X A^T) via fp32 WMMA 16x16x4,
// plus hyperbolic-MLR epilogue.
// Block = 256 threads = 8 wave32s arranged 2 (M) x 4 (N); each wave owns one
// 16x16 output tile -> block tile = 32 x 64.
// ---------------------------------------------------------------------------
__global__ __launch_bounds__(256) void hyp_mlr_kernel(
    const float* __restrict__ x, const float* __restrict__ p,
    const float* __restrict__ a, const float* __restrict__ ws,
    float* __restrict__ out) {
  __shared__ float Xs[TM * LSTR];  // 8704 B
  __shared__ float Ps[TN * LSTR];  // 17408 B
  __shared__ float As[TN * LSTR];  // 17408 B  (total 43520 B static LDS)

  const int tid = threadIdx.x;
  const int kBase = blockIdx.x * TN;
  const int bBase = blockIdx.y * TM;

  const int lane = tid & 31;
  const int wave = tid >> 5;
  const int mBase = (wave >> 2) * 16;  // 0 or 16
  const int nBase = (wave & 3) * 16;   // 0,16,32,48
  const int lmod = lane & 15;
  const int lh = lane >> 4;  // half-wave: selects K pair {2h, 2h+1}

  v8f accP = {};  // <x_b, p_k> accumulator (16x16 f32 C/D layout)
  v8f accA = {};  // <x_b, a_k> accumulator

  const int c4 = (tid & 15) * 4;        // float4 column within chunk
  const int row0 = tid >> 4;            // 16 rows covered per pass

  // Per-wave LDS fragment base pointers (A-frag from Xs, B-frags from Ps/As).
  const float* xrow = &Xs[(mBase + lmod) * LSTR + 2 * lh];
  const float* prow = &Ps[(nBase + lmod) * LSTR + 2 * lh];
  const float* arow = &As[(nBase + lmod) * LSTR + 2 * lh];

  for (int d0 = 0; d0 < Ddim; d0 += DC) {
    // ---- cooperative stage: X tile (TM x DC), P/A tiles (TN x DC) ----
#pragma unroll
    for (int r = row0; r < TM; r += 16) {
      const float4 v = *reinterpret_cast<const float4*>(
          x + (size_t)(bBase + r) * Ddim + d0 + c4);
      *reinterpret_cast<float4*>(&Xs[r * LSTR + c4]) = v;
    }
#pragma unroll
    for (int r = row0; r < TN; r += 16) {
      const float4 vp = *reinterpret_cast<const float4*>(
          p + (size_t)(kBase + r) * Ddim + d0 + c4);
      const float4 va = *reinterpret_cast<const float4*>(
          a + (size_t)(kBase + r) * Ddim + d0 + c4);
      *reinterpret_cast<float4*>(&Ps[r * LSTR + c4]) = vp;
      *reinterpret_cast<float4*>(&As[r * LSTR + c4]) = va;
    }
    __syncthreads();

    // ---- 16 K-steps of V_WMMA_F32_16X16X4_F32, X fragment shared ----
#pragma unroll
    for (int d = 0; d < DC; d += 4) {
      const v2f af = *reinterpret_cast<const v2f*>(xrow + d);
      const v2f bp = *reinterpret_cast<const v2f*>(prow + d);
      const v2f ba = *reinterpret_cast<const v2f*>(arow + d);
      accP = __builtin_amdgcn_wmma_f32_16x16x4_f32(
          false, af, false, bp, (short)0, accP, false, false);
      accA = __builtin_amdgcn_wmma_f32_16x16x4_f32(
          false, af, false, ba, (short)0, accA, false, false);
    }
    __syncthreads();
  }

  // ---- epilogue: hyperbolic MLR logit from the two dot-product planes ----
  const float* wp2 = ws;
  const float* wpa = ws + Kdim;
  const float* wna = ws + 2 * Kdim;
  const float* wln = ws + 3 * Kdim;
  const float* wx2 = ws + 4 * Kdim;

  const int k = kBase + nBase + lmod;        // this lane's output column
  const int bRow0 = bBase + mBase + 8 * lh;  // first of 8 output rows

  const float p2k = wp2[k];
  const float pak = wpa[k];
  const float nak = wna[k];
  const float lamnak = wln[k];
  const float t = 1.0f - p2k;  // beta = 1 - ||p||^2

#pragma unroll
  for (int r = 0; r < 8; ++r) {
    const int b = bRow0 + r;
    const float px = accP[r];  // <x_b, p_k>
    const float xa = accA[r];  // <x_b, a_k>
    const float v2 = wx2[b];   // ||x_b||^2

    const float uv = -px;                               // <u,v>, u = -p
    const float alpha = 1.0f + 2.0f * uv + v2;          // coeff of u
    const float gamma = 1.0f + 2.0f * uv + p2k * v2 + EPS;
    const float invg = 1.0f / gamma;
    const float z2 =
        (alpha * alpha * p2k + 2.0f * alpha * t * uv + t * t * v2) * invg * invg;
    const float za = t * (t * xa - alpha * pak) * invg;  // <z, a_pt>
    const float arg = 2.0f * za / ((1.0f - z2) * nak + EPS);
    out[(size_t)b * Kdim + k] = lamnak * asinhf(arg);
  }
}

// ---------------------------------------------------------------------------
extern "C" void kernel_launch(void* const* d_in, const int* in_sizes, int n_in,
                              void* d_out, int out_size, void* d_ws,
                              size_t ws_size, hipStream_t stream) {
  (void)in_sizes; (void)n_in; (void)out_size; (void)ws_size;
  const float* x = (const float*)d_in[0];
  const float* p = (const float*)d_in[1];
  const float* a = (const float*)d_in[2];
  float* out = (float*)d_out;
  float* ws = (float*)d_ws;  // needs (4*K + B) * 4 = 18 KiB

  hipLaunchKernelGGL(hyp_stats_kernel, dim3(Kdim + Bdim), dim3(Ddim), 0, stream,
                     x, p, a, ws);
  hipLaunchKernelGGL(hyp_mlr_kernel, dim3(Kdim / TN, Bdim / TM), dim3(256), 0,
                     stream, x, p, a, ws, out);
}